// PartialMatchingLoss_20409684590743
// MI455X (gfx1250) — compile-verified
//
#include <hip/hip_runtime.h>
#include <cmath>

typedef __attribute__((ext_vector_type(2))) float v2f;
typedef __attribute__((ext_vector_type(8))) float v8f;

#define BIGF 3.402823466e38f

constexpr int B  = 8;
constexpr int N  = 8192;   // completed points per batch
constexpr int M  = 4096;   // partial points per batch
constexpr int NT = 1024;   // completed points per LDS tile (16 KB as float4)
constexpr int WAVES_PER_BLOCK = 8;
constexpr int MT = WAVES_PER_BLOCK * 16;            // 128 partial rows per block
constexpr int BLOCKS = B * (M / MT);                // 256
constexpr int TOTAL_WAVES = BLOCKS * WAVES_PER_BLOCK; // 2048

__global__ __launch_bounds__(256)
void pml_min_kernel(const float* __restrict__ completed,
                    const float* __restrict__ partial,
                    float* __restrict__ wave_sums)
{
    __shared__ float4 tile[NT];   // {x, y, z, |c|^2} per completed point

    const int tid  = threadIdx.x;
    const int wave = tid >> 5;
    const int lane = tid & 31;
    const int hi   = lane >> 4;   // 0: holds K=0,1 / rows 0..7 of C; 1: K=2,3 / rows 8..15
    const int l16  = lane & 15;

    const int b      = blockIdx.x / (M / MT);
    const int mtile  = blockIdx.x % (M / MT);
    const int m_base = mtile * MT + wave * 16;

    // ---- A fragment (held for entire N loop): row (lane&15) = (-2px,-2py,-2pz,1) ----
    const float* prow = partial + ((size_t)b * M + m_base + l16) * 3;
    const float px = prow[0], py = prow[1], pz = prow[2];
    const float psq = px * px + py * py + pz * pz;
    v2f a;
    if (hi == 0) { a.x = -2.0f * px; a.y = -2.0f * py; }
    else         { a.x = -2.0f * pz; a.y = 1.0f;       }

    // ---- C fragment: cinit[i] = |p|^2 of row (i + 8*hi), broadcast along lanes (N dim) ----
    v8f cinit;
    #pragma unroll
    for (int i = 0; i < 8; ++i)
        cinit[i] = __shfl(psq, i + 8 * hi, 32);

    float dmin[8];
    #pragma unroll
    for (int i = 0; i < 8; ++i) dmin[i] = BIGF;

    const float* cbase = completed + (size_t)b * N * 3;

    for (int t0 = 0; t0 < N; t0 += NT) {
        // ---- cooperative stage of NT completed points into LDS ----
        #pragma unroll
        for (int j = 0; j < 4; ++j) {
            const int idx = tid * 4 + j;
            const float* c = cbase + (size_t)(t0 + idx) * 3;
            const float x = c[0], y = c[1], z = c[2];
            tile[idx] = make_float4(x, y, z, x * x + y * y + z * z);
        }
        if (t0 + NT < N)   // warm next tile into cache (global_prefetch_b8)
            __builtin_prefetch(cbase + (size_t)(t0 + NT + tid * 4) * 3, 0, 0);
        __syncthreads();

        // ---- 64 WMMAs per tile: D = psq + csq - 2 p.c = d^2, then running min ----
        for (int c16 = 0; c16 < NT; c16 += 16) {
            // lanes 0-15 read {x,y}, lanes 16-31 read {z,|c|^2}: one 8B-aligned b64, conflict-free
            const float* p = reinterpret_cast<const float*>(&tile[c16 + l16]) + (hi ? 2 : 0);
            const float2 bv = *reinterpret_cast<const float2*>(p);
            v2f bf; bf.x = bv.x; bf.y = bv.y;
            v8f d = __builtin_amdgcn_wmma_f32_16x16x4_f32(
                        false, a, false, bf, (short)0, cinit, false, false);
            #pragma unroll
            for (int i = 0; i < 8; ++i)
                dmin[i] = fminf(dmin[i], d[i]);
        }
        __syncthreads();
    }

    // ---- reduce min across the 16 lanes (N residues) per row, then sum sqrt per wave ----
    float s = 0.0f;
    #pragma unroll
    for (int i = 0; i < 8; ++i) {
        float v = dmin[i];
        v = fminf(v, __shfl_xor(v, 1, 32));
        v = fminf(v, __shfl_xor(v, 2, 32));
        v = fminf(v, __shfl_xor(v, 4, 32));
        v = fminf(v, __shfl_xor(v, 8, 32));
        s += sqrtf(fmaxf(v, 0.0f));   // rows i+8*hi covered per half-wave
    }
    s += __shfl_xor(s, 16, 32);       // combine the two half-wave row groups
    if (lane == 0)
        wave_sums[blockIdx.x * WAVES_PER_BLOCK + wave] = s;
}

__global__ __launch_bounds__(256)
void pml_reduce_kernel(const float* __restrict__ wave_sums, float* __restrict__ out)
{
    __shared__ float sm[256];
    const int t = threadIdx.x;
    float s = 0.0f;
    for (int i = t; i < TOTAL_WAVES; i += 256) s += wave_sums[i];
    sm[t] = s;
    __syncthreads();
    for (int off = 128; off > 0; off >>= 1) {
        if (t < off) sm[t] += sm[t + off];
        __syncthreads();
    }
    if (t == 0) out[0] = sm[0] / (float)(B * M);
}

extern "C" void kernel_launch(void* const* d_in, const int* in_sizes, int n_in,
                              void* d_out, int out_size, void* d_ws, size_t ws_size,
                              hipStream_t stream)
{
    const float* completed = (const float*)d_in[0];   // (8, 8192, 3) f32
    const float* partial   = (const float*)d_in[1];   // (8, 4096, 3) f32
    float* wave_sums = (float*)d_ws;                  // 2048 floats of scratch

    pml_min_kernel<<<BLOCKS, 256, 0, stream>>>(completed, partial, wave_sums);
    pml_reduce_kernel<<<1, 256, 0, stream>>>(wave_sums, (float*)d_out);
}